// MACELayer_79499844649453
// MI455X (gfx1250) — compile-verified
//
#include <hip/hip_runtime.h>

// MACE layer for MI455X (gfx1250, wave32, WMMA).
// Radial MLP = fp32 GEMM on the matrix pipe via V_WMMA_F32_16X16X4_F32,
// fused with the channelwise tensor product + atomic scatter so the
// [E,5,64] edge-weight tensor never touches HBM.

#define NN   32768
#define EE   262144
#define FF   64
#define ZZ   10
#define RR   8
#define HH   64
#define PTP  5
#define WCOLS (PTP * FF)   // 320

typedef __attribute__((ext_vector_type(2))) float v2f;
typedef __attribute__((ext_vector_type(8))) float v8f;

// Fast SiLU: one v_exp + one v_rcp (TRANS pipe) instead of the IEEE
// div_scale/rcp/fma fixup chain the plain division lowers to.
__device__ __forceinline__ float silu_f(float x) {
  return x * __builtin_amdgcn_rcpf(1.0f + __expf(-x));
}

// 12-byte packed vector gather (lowers to global_load_b96).
struct __attribute__((packed, aligned(4))) F3 { float x, y, z; };

// One 16(M=edges) x outN GEMM layer, K accumulated in steps of 4 via
// V_WMMA_F32_16X16X4_F32.  Fragment layouts per CDNA5 ISA 7.12.2:
//   A 16x4 f32 : lane m16 holds row M=m16; VGPR0/1 = K = {0,1} (lanes 0-15)
//                or K = {2,3} (lanes 16-31)
//   B 4x16 f32 : VGPR0/1 = rows K = {0,1} | {2,3} by lane half, N = lane%16
//   C/D 16x16  : VGPR r -> M = r + 8*(lane/16), N = lane%16
__device__ __forceinline__ void mlp_layer_wmma(
    const float* __restrict__ X,   // LDS activations, stride 64, [16][inK]
    const float* __restrict__ W,   // global weights, row-major [inK][outN]
    float* __restrict__ Y,         // LDS output, stride ystride
    int inK, int outN, int ystride, float scale, bool act,
    int m16, int hi)
{
  const int ka = hi * 2;
  for (int j = 0; j < outN; j += 16) {
    v8f c = {};
    for (int kk = 0; kk < inK; kk += 4) {
      v2f a, b;
      a.x = X[m16 * 64 + kk + ka];
      a.y = X[m16 * 64 + kk + ka + 1];
      b.x = W[(size_t)(kk + ka) * outN + j + m16];
      b.y = W[(size_t)(kk + ka + 1) * outN + j + m16];
      c = __builtin_amdgcn_wmma_f32_16x16x4_f32(
              /*neg_a=*/false, a, /*neg_b=*/false, b,
              /*c_mod=*/(short)0, c, /*reuse_a=*/false, /*reuse_b=*/false);
    }
#pragma unroll
    for (int r = 0; r < 8; ++r) {
      float y = c[r] * scale;
      if (act) y = silu_f(y);
      Y[(size_t)(r + 8 * hi) * ystride + j + m16] = y;
    }
  }
}

// ---------------------------------------------------------------------------
// Kernel 1: per-node prep — per-species skip linear (linZ) + linear_up.
// Block = 4 nodes x 64 output channels.
// ---------------------------------------------------------------------------
__global__ __launch_bounds__(256) void node_prep_kernel(
    const float* __restrict__ node_feats, const int* __restrict__ species,
    const float* __restrict__ Wss, const float* __restrict__ Wsv,
    const float* __restrict__ Wus, const float* __restrict__ Wuv,
    float* __restrict__ s_up, float* __restrict__ v_up,
    float* __restrict__ skip_s, float* __restrict__ skip_v)
{
  __shared__ float sh_s[4][64];
  __shared__ float sh_v[4][3][64];

  const int t  = threadIdx.x;
  const int nl = t >> 6;
  const int g  = t & 63;
  const int n  = blockIdx.x * 4 + nl;

  const float* nf = node_feats + (size_t)n * (4 * FF);
  sh_s[nl][g]    = nf[g];
  sh_v[nl][0][g] = nf[FF + g * 3 + 0];
  sh_v[nl][1][g] = nf[FF + g * 3 + 1];
  sh_v[nl][2][g] = nf[FF + g * 3 + 2];
  __syncthreads();

  const int spec = species[n];
  const float* Ws = Wss + (size_t)spec * FF * FF;
  const float* Wv = Wsv + (size_t)spec * FF * FF;

  float aSS = 0.f, aS0 = 0.f, aS1 = 0.f, aS2 = 0.f;
  float aUS = 0.f, aU0 = 0.f, aU1 = 0.f, aU2 = 0.f;
#pragma unroll 4
  for (int f = 0; f < FF; ++f) {
    const float sf = sh_s[nl][f];
    const float v0 = sh_v[nl][0][f], v1 = sh_v[nl][1][f], v2 = sh_v[nl][2][f];
    const float wks = Ws[f * FF + g], wkv = Wv[f * FF + g];
    const float wus = Wus[f * FF + g], wuv = Wuv[f * FF + g];
    aSS += sf * wks;
    aS0 += v0 * wkv;  aS1 += v1 * wkv;  aS2 += v2 * wkv;
    aUS += sf * wus;
    aU0 += v0 * wuv;  aU1 += v1 * wuv;  aU2 += v2 * wuv;
  }

  const float invz = 0.0395284707521f;  // 1/sqrt(F*Z) = 1/sqrt(640)
  const float invf = 0.125f;            // 1/sqrt(64)
  const size_t base = (size_t)n * FF + g;
  skip_s[base] = aSS * invz;
  skip_v[base * 3 + 0] = aS0 * invz;
  skip_v[base * 3 + 1] = aS1 * invz;
  skip_v[base * 3 + 2] = aS2 * invz;
  s_up[base] = aUS * invf;
  v_up[base * 3 + 0] = aU0 * invf;
  v_up[base * 3 + 1] = aU1 * invf;
  v_up[base * 3 + 2] = aU2 * invf;
}

// ---------------------------------------------------------------------------
// Kernel 2: fused per-edge pipeline. One wave (32 threads) = 16 edges.
//   radial MLP (WMMA fp32) -> path weights in LDS -> tensor product ->
//   atomic scatter into receiver accumulators.
// LDS/block ~28.5 KB -> ~11 blocks per 320 KB WGP.
// ---------------------------------------------------------------------------
__global__ __launch_bounds__(32) void edge_tp_kernel(
    const float* __restrict__ vectors,
    const float* __restrict__ re,
    const float* __restrict__ W1, const float* __restrict__ W2,
    const float* __restrict__ W3, const float* __restrict__ W4,
    const float* __restrict__ s_up, const float* __restrict__ v_up,
    const int* __restrict__ senders, const int* __restrict__ receivers,
    float* __restrict__ agg_s, float* __restrict__ agg_v)
{
  __shared__ float Xa[16 * 64];
  __shared__ float Xb[16 * 64];
  __shared__ float Wl[16 * WCOLS];
  __shared__ float Yv[16 * 3];

  const int lane = threadIdx.x;
  const int m16  = lane & 15;
  const int hi   = lane >> 4;
  const int e0   = blockIdx.x * 16;

  __builtin_prefetch(W2, 0, 0);
  __builtin_prefetch(W4, 0, 0);

  // stage radial embeddings [16][8] into activation tile (stride 64)
  for (int i = lane; i < 16 * RR; i += 32) {
    const int e = i >> 3, k = i & 7;
    Xa[e * 64 + k] = re[(size_t)(e0 + e) * RR + k];
  }
  // spherical harmonics l=1: unit edge vectors
  if (lane < 16) {
    const float x = vectors[(size_t)(e0 + lane) * 3 + 0];
    const float y = vectors[(size_t)(e0 + lane) * 3 + 1];
    const float z = vectors[(size_t)(e0 + lane) * 3 + 2];
    const float inv = __builtin_amdgcn_rcpf(sqrtf(x * x + y * y + z * z) + 1e-9f);
    Yv[lane * 3 + 0] = x * inv;
    Yv[lane * 3 + 1] = y * inv;
    Yv[lane * 3 + 2] = z * inv;
  }
  __syncthreads();

  // radial MLP: [16,8] -> 64 -> 64 -> 64 -> 320, silu between, fan-in scaled
  mlp_layer_wmma(Xa, W1, Xb, RR, HH, 64,    0.35355339059f, true,  m16, hi);
  __syncthreads();
  mlp_layer_wmma(Xb, W2, Xa, HH, HH, 64,    0.125f,         true,  m16, hi);
  __syncthreads();
  mlp_layer_wmma(Xa, W3, Xb, HH, HH, 64,    0.125f,         true,  m16, hi);
  __syncthreads();
  mlp_layer_wmma(Xb, W4, Wl, HH, WCOLS, WCOLS, 0.125f,      false, m16, hi);
  __syncthreads();

  // channelwise tensor product + scatter. 16 edges x 64 ch = 1024 pairs.
  const float EPS = 0.25f;
  const float inv_sqrt2 = 0.70710678118f;
#pragma unroll 4
  for (int it = 0; it < 32; ++it) {
    const int idx = it * 32 + lane;
    const int e = idx >> 6;
    const int c = idx & 63;
    const int ge  = e0 + e;
    const int snd = senders[ge];
    const int rcv = receivers[ge];
    const float yx = Yv[e * 3 + 0], yy = Yv[e * 3 + 1], yz = Yv[e * 3 + 2];

    const size_t sb = (size_t)snd * FF + c;
    const float ss = s_up[sb];
    const F3 vv = *(const F3*)(v_up + sb * 3);   // global_load_b96
    const float vx = vv.x, vy = vv.y, vz = vv.z;

    const float* we = Wl + e * WCOLS + c;
    const float w0 = we[0], w1 = we[64], w2 = we[128], w3 = we[192], w4 = we[256];

    const float dot = vx * yx + vy * yy + vz * yz;
    const float cx = vy * yz - vz * yy;
    const float cy = vz * yx - vx * yz;
    const float cz = vx * yy - vy * yx;

    const float ms = (w0 * ss + w3 * dot) * EPS;
    const float f4 = w4 * inv_sqrt2;
    const float mvx = (w1 * yx + w2 * vx + f4 * cx) * EPS;
    const float mvy = (w1 * yy + w2 * vy + f4 * cy) * EPS;
    const float mvz = (w1 * yz + w2 * vz + f4 * cz) * EPS;

    const size_t rb = (size_t)rcv * FF + c;
    atomicAdd(&agg_s[rb], ms);
    atomicAdd(&agg_v[rb * 3 + 0], mvx);
    atomicAdd(&agg_v[rb * 3 + 1], mvy);
    atomicAdd(&agg_v[rb * 3 + 2], mvz);
  }
}

// ---------------------------------------------------------------------------
// Kernel 3: node finalize — linear_down, symmetric contraction, linear_post,
// skip add, merged write, scalar readout. Block = 4 nodes x 64 channels.
// ---------------------------------------------------------------------------
__global__ __launch_bounds__(256) void node_finalize_kernel(
    const float* __restrict__ agg_s, const float* __restrict__ agg_v,
    const float* __restrict__ skip_s, const float* __restrict__ skip_v,
    const int* __restrict__ species,
    const float* __restrict__ Wds, const float* __restrict__ Wdv,
    const float* __restrict__ Wsc,
    const float* __restrict__ Wps, const float* __restrict__ Wpv,
    const float* __restrict__ Wout,
    float* __restrict__ out)
{
  __shared__ float sh_s[4][64];
  __shared__ float sh_v[4][3][64];
  __shared__ float sh_so[4][64];
  __shared__ float sh_vo[4][3][64];
  __shared__ float red[4][64];

  const int t  = threadIdx.x;
  const int nl = t >> 6;
  const int g  = t & 63;
  const int n  = blockIdx.x * 4 + nl;
  const size_t base = (size_t)n * FF + g;

  sh_s[nl][g] = agg_s[base];
  const F3 av = *(const F3*)(agg_v + base * 3);
  sh_v[nl][0][g] = av.x;
  sh_v[nl][1][g] = av.y;
  sh_v[nl][2][g] = av.z;
  __syncthreads();

  // linear_down
  float ds = 0.f, dv0 = 0.f, dv1 = 0.f, dv2 = 0.f;
#pragma unroll 4
  for (int f = 0; f < FF; ++f) {
    const float ws_ = Wds[f * FF + g];
    const float wv_ = Wdv[f * FF + g];
    ds  += sh_s[nl][f]    * ws_;
    dv0 += sh_v[nl][0][f] * wv_;
    dv1 += sh_v[nl][1][f] * wv_;
    dv2 += sh_v[nl][2][f] * wv_;
  }
  ds *= 0.125f; dv0 *= 0.125f; dv1 *= 0.125f; dv2 *= 0.125f;

  // per-species symmetric contraction (correlation 1..3)
  const int spec = species[n];
  const float* wz = Wsc + (size_t)spec * 9 * FF + g;
  const float vn2 = dv0 * dv0 + dv1 * dv1 + dv2 * dv2;
  const float s2  = ds * ds;
  const float so = wz[0 * FF] * ds + wz[1 * FF] * s2 + wz[2 * FF] * vn2
                 + wz[3 * FF] * s2 * ds + wz[4 * FF] * ds * vn2;
  const float gv = wz[5 * FF] + wz[6 * FF] * ds + wz[7 * FF] * s2 + wz[8 * FF] * vn2;
  sh_so[nl][g]    = so;
  sh_vo[nl][0][g] = gv * dv0;
  sh_vo[nl][1][g] = gv * dv1;
  sh_vo[nl][2][g] = gv * dv2;
  __syncthreads();

  // linear_post
  float ps = 0.f, pv0 = 0.f, pv1 = 0.f, pv2 = 0.f;
#pragma unroll 4
  for (int f = 0; f < FF; ++f) {
    const float wp = Wps[f * FF + g];
    const float wq = Wpv[f * FF + g];
    ps  += sh_so[nl][f]    * wp;
    pv0 += sh_vo[nl][0][f] * wq;
    pv1 += sh_vo[nl][1][f] * wq;
    pv2 += sh_vo[nl][2][f] * wq;
  }
  ps  = ps  * 0.125f + skip_s[base];
  pv0 = pv0 * 0.125f + skip_v[base * 3 + 0];
  pv1 = pv1 * 0.125f + skip_v[base * 3 + 1];
  pv2 = pv2 * 0.125f + skip_v[base * 3 + 2];

  // merged output [N, 4F] at d_out + N
  float* mg = out + NN + (size_t)n * (4 * FF);
  mg[g] = ps;
  mg[FF + g * 3 + 0] = pv0;
  mg[FF + g * 3 + 1] = pv1;
  mg[FF + g * 3 + 2] = pv2;

  // readout: out[n] = (1/8) * sum_g ps[g] * Wout[g]
  red[nl][g] = ps * Wout[g];
  __syncthreads();
  for (int off = 32; off > 0; off >>= 1) {
    if (g < off) red[nl][g] += red[nl][g + off];
    __syncthreads();
  }
  if (g == 0) out[n] = red[nl][0] * 0.125f;
}

__global__ void zero_kernel(float* __restrict__ p, size_t n) {
  const size_t i = (size_t)blockIdx.x * blockDim.x + threadIdx.x;
  if (i < n) p[i] = 0.f;
}

// ---------------------------------------------------------------------------
extern "C" void kernel_launch(void* const* d_in, const int* in_sizes, int n_in,
                              void* d_out, int out_size, void* d_ws, size_t ws_size,
                              hipStream_t stream) {
  const float* vectors    = (const float*)d_in[0];
  const float* node_feats = (const float*)d_in[1];
  const float* re         = (const float*)d_in[2];
  const float* W_skip_s   = (const float*)d_in[3];
  const float* W_skip_v   = (const float*)d_in[4];
  const float* W_up_s     = (const float*)d_in[5];
  const float* W_up_v     = (const float*)d_in[6];
  const float* W_mlp1     = (const float*)d_in[7];
  const float* W_mlp2     = (const float*)d_in[8];
  const float* W_mlp3     = (const float*)d_in[9];
  const float* W_mlp4     = (const float*)d_in[10];
  const float* W_down_s   = (const float*)d_in[11];
  const float* W_down_v   = (const float*)d_in[12];
  const float* W_sc       = (const float*)d_in[13];
  const float* W_post_s   = (const float*)d_in[14];
  const float* W_post_v   = (const float*)d_in[15];
  const float* W_out      = (const float*)d_in[16];
  const int*   species    = (const int*)d_in[17];
  const int*   senders    = (const int*)d_in[18];
  const int*   receivers  = (const int*)d_in[19];
  float* out = (float*)d_out;

  // workspace layout (floats): 3 * N * 256 = ~100.7 MB
  float* ws     = (float*)d_ws;
  float* s_up   = ws;
  float* v_up   = s_up   + (size_t)NN * FF;
  float* skip_s = v_up   + (size_t)NN * FF * 3;
  float* skip_v = skip_s + (size_t)NN * FF;
  float* agg_s  = skip_v + (size_t)NN * FF * 3;
  float* agg_v  = agg_s  + (size_t)NN * FF;       // contiguous after agg_s

  const size_t zn = (size_t)NN * (4 * FF);        // agg_s + agg_v together
  zero_kernel<<<(unsigned)((zn + 255) / 256), 256, 0, stream>>>(agg_s, zn);

  node_prep_kernel<<<NN / 4, 256, 0, stream>>>(
      node_feats, species, W_skip_s, W_skip_v, W_up_s, W_up_v,
      s_up, v_up, skip_s, skip_v);

  edge_tp_kernel<<<EE / 16, 32, 0, stream>>>(
      vectors, re, W_mlp1, W_mlp2, W_mlp3, W_mlp4,
      s_up, v_up, senders, receivers, agg_s, agg_v);

  node_finalize_kernel<<<NN / 4, 256, 0, stream>>>(
      agg_s, agg_v, skip_s, skip_v, species,
      W_down_s, W_down_v, W_sc, W_post_s, W_post_v, W_out, out);
}